// RetAttention_38517266710797
// MI455X (gfx1250) — compile-verified
//
#include <hip/hip_runtime.h>

typedef __attribute__((ext_vector_type(2))) float v2f;
typedef __attribute__((ext_vector_type(8))) float v8f;
typedef __attribute__((address_space(3))) float lds_f;

#define B_  2
#define H_  16
#define S_  2048
#define D_  128
#define TC  16                 // key-chunk length
#define WAVES 4                // waves per block
#define QROWS_PER_WAVE 16
#define QROWS_PER_BLOCK (WAVES * QROWS_PER_WAVE)   // 64
#define KSTRIDE 132            // padded LDS row stride (floats) -> bank-conflict free
#define PSTRIDE 17             // padded P-scratch row stride

// Direct cache->LDS async copy (ASYNCcnt-tracked), bypassing VGPR staging.
__device__ __forceinline__ void async_ld_b128(lds_f* lds_dst, const float* gsrc) {
    asm volatile("global_load_async_to_lds_b128 %0, %1, off"
                 :: "v"(lds_dst), "v"(gsrc) : "memory");
}

__launch_bounds__(WAVES * 32, 1)
__global__ void RetAttention_38517266710797_kernel(
    const float* __restrict__ q, const float* __restrict__ k,
    const float* __restrict__ v, const float* __restrict__ mask,
    float* __restrict__ o)
{
    __shared__ float Klds[2][TC * KSTRIDE];         // double-buffered, 2 x 8448 B
    __shared__ float Vlds[2][TC * KSTRIDE];         // double-buffered, 2 x 8448 B
    __shared__ float Plds[WAVES][16 * PSTRIDE];     // 4352 B

    const int tid  = threadIdx.x;     // 0..127
    const int wave = tid >> 5;
    const int lane = tid & 31;
    const int ln   = lane & 15;       // n (col) / m (row) slot
    const int hi   = lane >> 4;       // half-wave selector

    const int bh    = blockIdx.y;                 // 0..B*H-1
    const int h     = bh % H_;
    const int qrow0 = blockIdx.x * QROWS_PER_BLOCK + wave * QROWS_PER_WAVE;

    const float* qb = q    + (size_t)bh * S_ * D_;
    const float* kb = k    + (size_t)bh * S_ * D_;
    const float* vb = v    + (size_t)bh * S_ * D_;
    const float* mb = mask + (size_t)h  * S_ * S_;
    float*       ob = o    + (size_t)bh * S_ * D_;

    // Per-thread K/V staging slots (4 float4 each for K and V): 8 async ops/wave/chunk.
    int srow[4], scol[4];
    #pragma unroll
    for (int i = 0; i < 4; ++i) {
        const int idx = tid + i * 128;   // 0..511 float4 slots
        srow[i] = idx >> 5;              // 0..15
        scol[i] = (idx & 31) * 4;        // float column 0..124
    }

    // ---- Prologue: kick off async staging of chunk 0 into buffer 0
    #pragma unroll
    for (int i = 0; i < 4; ++i) {
        async_ld_b128((lds_f*)&Klds[0][srow[i] * KSTRIDE + scol[i]],
                      kb + (size_t)srow[i] * D_ + scol[i]);
        async_ld_b128((lds_f*)&Vlds[0][srow[i] * KSTRIDE + scol[i]],
                      vb + (size_t)srow[i] * D_ + scol[i]);
    }

    // ---- Preload the Q tile as WMMA A-fragments (16x4 per step, 32 steps over D=128)
    v2f qfrag[32];
    {
        const float* qrow = qb + (size_t)(qrow0 + ln) * D_;
        #pragma unroll
        for (int kk = 0; kk < 32; ++kk) {
            const int kd = kk * 4 + hi * 2;
            qfrag[kk].x = qrow[kd + 0];
            qfrag[kk].y = qrow[kd + 1];
        }
    }

    v8f acc[8];
    #pragma unroll
    for (int nt = 0; nt < 8; ++nt) acc[nt] = v8f{};
    float rsum[8];
    #pragma unroll
    for (int j = 0; j < 8; ++j) rsum[j] = 0.0f;

    for (int t0 = 0, it = 0; t0 < S_; t0 += TC, ++it) {
        const int buf = it & 1;

        // ---- Mask tile for this wave's rows (C/D layout: row j+8*hi, col ln)
        float mfrag[8];
        #pragma unroll
        for (int j = 0; j < 8; ++j) {
            const int row = qrow0 + j + 8 * hi;
            mfrag[j] = mb[(size_t)row * S_ + (t0 + ln)];
        }

        __syncthreads();   // everyone done READING buf^1 -> safe to overwrite it

        // ---- Issue async staging for the NEXT chunk into buf^1 (overlaps compute)
        const int tn = (t0 + TC < S_) ? (t0 + TC) : 0;   // tail: harmless reload
        #pragma unroll
        for (int i = 0; i < 4; ++i) {
            async_ld_b128((lds_f*)&Klds[buf ^ 1][srow[i] * KSTRIDE + scol[i]],
                          kb + (size_t)(tn + srow[i]) * D_ + scol[i]);
            async_ld_b128((lds_f*)&Vlds[buf ^ 1][srow[i] * KSTRIDE + scol[i]],
                          vb + (size_t)(tn + srow[i]) * D_ + scol[i]);
        }

        // Async loads complete in issue order: <=8 outstanding means the 8 ops
        // of the CURRENT chunk (issued last iteration / prologue) have landed.
        asm volatile("s_wait_asynccnt 0x8" ::: "memory");
        __syncthreads();   // all waves' chunk-i staging visible

        // ---- 16x16 score tile: P = Q_tile * K_chunk^T, two interleaved WMMA chains
        v8f p0 = v8f{}, p1 = v8f{};
        #pragma unroll
        for (int kk = 0; kk < 32; kk += 2) {
            v2f b0, b1;
            const int kd0 = kk * 4 + hi * 2;
            b0.x = Klds[buf][ln * KSTRIDE + kd0 + 0];
            b0.y = Klds[buf][ln * KSTRIDE + kd0 + 1];
            b1.x = Klds[buf][ln * KSTRIDE + kd0 + 4];
            b1.y = Klds[buf][ln * KSTRIDE + kd0 + 5];
            p0 = __builtin_amdgcn_wmma_f32_16x16x4_f32(
                     false, qfrag[kk + 0], false, b0, (short)0, p0, false, false);
            p1 = __builtin_amdgcn_wmma_f32_16x16x4_f32(
                     false, qfrag[kk + 1], false, b1, (short)0, p1, false, false);
        }

        // ---- mask multiply + |.| row-sum accumulation; stash P for C->A relayout
        #pragma unroll
        for (int j = 0; j < 8; ++j) {
            const float pm = (p0[j] + p1[j]) * mfrag[j];
            rsum[j] += __builtin_fabsf(pm);
            Plds[wave][(j + 8 * hi) * PSTRIDE + ln] = pm;
        }
        // Same-wave DS ops retire in order; drain before the A-fragment reads.
        asm volatile("s_wait_dscnt 0x0" ::: "memory");

        // ---- O += P * V_chunk : P is a 16x16 A-matrix, V is 16x128 (8 N-tiles)
        #pragma unroll
        for (int nt = 0; nt < 8; ++nt) {
            v8f c = acc[nt];
            #pragma unroll
            for (int kk2 = 0; kk2 < 4; ++kk2) {
                const int kcol = kk2 * 4 + hi * 2;
                v2f afrag, bfrag;
                afrag.x = Plds[wave][ln * PSTRIDE + kcol + 0];
                afrag.y = Plds[wave][ln * PSTRIDE + kcol + 1];
                bfrag.x = Vlds[buf][(kcol + 0) * KSTRIDE + nt * 16 + ln];
                bfrag.y = Vlds[buf][(kcol + 1) * KSTRIDE + nt * 16 + ln];
                c = __builtin_amdgcn_wmma_f32_16x16x4_f32(
                        false, afrag, false, bfrag, (short)0, c, false, false);
            }
            acc[nt] = c;
        }
    }

    // ---- Row-sum reduction across the 16 lanes sharing each score row
    #pragma unroll
    for (int j = 0; j < 8; ++j) {
        float s = rsum[j];
        s += __shfl_xor(s, 1, 32);
        s += __shfl_xor(s, 2, 32);
        s += __shfl_xor(s, 4, 32);
        s += __shfl_xor(s, 8, 32);
        rsum[j] = 1.0f / fmaxf(s, 1.0f);   // clip(.,1,inf) then reciprocal
    }

    // ---- Normalize + store (coalesced 64B per 16-lane group)
    #pragma unroll
    for (int nt = 0; nt < 8; ++nt) {
        #pragma unroll
        for (int j = 0; j < 8; ++j) {
            const int row = qrow0 + j + 8 * hi;
            ob[(size_t)row * D_ + nt * 16 + ln] = acc[nt][j] * rsum[j];
        }
    }
}

extern "C" void kernel_launch(void* const* d_in, const int* in_sizes, int n_in,
                              void* d_out, int out_size, void* d_ws, size_t ws_size,
                              hipStream_t stream) {
    const float* q    = (const float*)d_in[0];
    const float* k    = (const float*)d_in[1];
    const float* v    = (const float*)d_in[2];
    const float* mask = (const float*)d_in[3];
    float* o = (float*)d_out;

    dim3 grid(S_ / QROWS_PER_BLOCK, B_ * H_);   // (32, 32)
    RetAttention_38517266710797_kernel<<<grid, WAVES * 32, 0, stream>>>(q, k, v, mask, o);
}